// ParticleNet_21577915695780
// MI455X (gfx1250) — compile-verified
//
#include <hip/hip_runtime.h>

// ---------------------------------------------------------------------------
// ParticleNet forward for MI455X (gfx1250, wave32).
// All GEMM-shaped work (EdgeConv MLP layers, kNN Gram matrices) runs on the
// fp32 WMMA pipe (v_wmma_f32_16x16x4_f32), numerically equivalent to the fp32
// reference. All operands are staged in LDS so WMMA chains consume clustered
// ds_load_b64 fragments instead of serialized L2-latency global loads.
// ---------------------------------------------------------------------------

#define EPS 1e-5f

constexpr int NB     = 512;   // graphs
constexpr int NN     = 256;   // nodes per graph
constexpr int FN     = 16;    // node features
constexpr int FG     = 4;     // graph features
constexpr int HID    = 64;
constexpr int NNODES = NB * NN;   // 131072

typedef __attribute__((ext_vector_type(2))) float v2f;
typedef __attribute__((ext_vector_type(8))) float v8f;

__device__ __forceinline__ v8f wmma4(v2f a, v2f b, v8f c) {
  // D = A(16x4,f32) * B(4x16,f32) + C(16x16,f32)
  return __builtin_amdgcn_wmma_f32_16x16x4_f32(false, a, false, b,
                                               (short)0, c, false, false);
}

// ---------------------------------------------------------------------------
// GraphNorm: per-graph, per-feature mean/var with learnable mean-scale alpha.
// ---------------------------------------------------------------------------
__global__ void graphnorm_k(const float* __restrict__ x,
                            const float* __restrict__ gamma,
                            const float* __restrict__ beta,
                            const float* __restrict__ alpha,
                            float* __restrict__ xn) {
  const int g = blockIdx.x;
  const int f = threadIdx.x;
  if (f >= FN) return;
  const float* xg = x + (size_t)g * NN * FN;
  float mean = 0.f;
  for (int n = 0; n < NN; ++n) mean += xg[n * FN + f];
  mean *= (1.f / NN);
  const float am = alpha[f] * mean;
  float var = 0.f;
  for (int n = 0; n < NN; ++n) {
    const float o = xg[n * FN + f] - am;
    var += o * o;
  }
  var *= (1.f / NN);
  const float sc = gamma[f] * rsqrtf(var + EPS);
  const float sh = beta[f];
  float* og = xn + (size_t)g * NN * FN;
  for (int n = 0; n < NN; ++n) og[n * FN + f] = (xg[n * FN + f] - am) * sc + sh;
}

// ---------------------------------------------------------------------------
// Per-graph kNN (k=4, loop=false). Block = 4 waves = half a graph (128 rows).
// The graph tile (256x64) is staged once into LDS (stride 68: float4-aligned,
// 4-bank row skew), squared norms are computed from LDS, then each wave builds
// two 16-row distance strips: per 16x16 tile, 16 B-fragments are loaded as a
// ds_load_b64 clause and drained by a 16-deep WMMA chain (A-frags live in
// VGPRs for the whole strip). Top-4 scan is wave-private (LDS is in-order).
// ---------------------------------------------------------------------------
__global__ __launch_bounds__(128)
void knn_k(const float* __restrict__ X, int* __restrict__ srcOut) {
  extern __shared__ float smem[];
  float* Xl   = smem;                     // [256][68]
  float* dstr = Xl + NN * 68;             // [4 waves][16][260]
  float* sqL  = dstr + 4 * 16 * 260;      // [256]
  const int lane = threadIdx.x & 31;
  const int wv   = threadIdx.x >> 5;
  const int lh   = lane >> 4;             // lane-half: 0 or 1
  const int lid  = lane & 15;
  const int g    = blockIdx.x >> 1;
  const int halfBase = (blockIdx.x & 1) * 128;
  const float* Xg = X + (size_t)g * NN * HID;

  // stage graph tile (coalesced float4)
  for (int idx = threadIdx.x; idx < NN * HID / 4; idx += 128) {
    const int row = idx >> 4;
    const int c4  = (idx & 15) << 2;
    *(float4*)(Xl + row * 68 + c4) = *(const float4*)(Xg + row * HID + c4);
  }
  __syncthreads();
  for (int r = threadIdx.x; r < NN; r += 128) {
    const float* p = Xl + r * 68;
    float s = 0.f;
    for (int k = 0; k < HID; ++k) s += p[k] * p[k];
    sqL[r] = s;
  }
  __syncthreads();

  float* drow = dstr + wv * 16 * 260;
  for (int i = 0; i < 2; ++i) {
    const int r0 = halfBase + (wv * 2 + i) * 16;
    v2f afr[16];
#pragma unroll
    for (int k4 = 0; k4 < 16; ++k4)
      afr[k4] = *(const v2f*)(Xl + (r0 + lid) * 68 + 4 * k4 + 2 * lh);
    for (int tc = 0; tc < 16; ++tc) {
      const int col0 = tc * 16;
      v2f bfr[16];
#pragma unroll
      for (int k4 = 0; k4 < 16; ++k4)
        bfr[k4] = *(const v2f*)(Xl + (col0 + lid) * 68 + 4 * k4 + 2 * lh);
      v8f acc = {};
#pragma unroll
      for (int k4 = 0; k4 < 16; ++k4) acc = wmma4(afr[k4], bfr[k4], acc);
      const int ncol = col0 + lid;
      const float sqn = sqL[ncol];
#pragma unroll
      for (int v = 0; v < 8; ++v) {
        const int m = v + 8 * lh;
        float d = sqL[r0 + m] + sqn - 2.f * acc[v];
        if (r0 + m == ncol) d += 1e9f;    // exclude self
        drow[m * 260 + ncol] = d;
      }
    }
    // top-4 insertion scan (strip is wave-private; DS ops are in-order)
    if (lane < 16) {
      const int row = lane;
      float bd0 = 3e38f, bd1 = 3e38f, bd2 = 3e38f, bd3 = 3e38f;
      int   bi0 = 0,     bi1 = 0,     bi2 = 0,     bi3 = 0;
      for (int c = 0; c < NN; ++c) {
        const float d = drow[row * 260 + c];
        if (d < bd3) {
          if (d < bd0)      { bd3=bd2;bi3=bi2; bd2=bd1;bi2=bi1; bd1=bd0;bi1=bi0; bd0=d;bi0=c; }
          else if (d < bd1) { bd3=bd2;bi3=bi2; bd2=bd1;bi2=bi1; bd1=d;bi1=c; }
          else if (d < bd2) { bd3=bd2;bi3=bi2; bd2=d;bi2=c; }
          else              { bd3=d;bi3=c; }
        }
      }
      const int node = g * NN + r0 + row;
      srcOut[node * 4 + 0] = g * NN + bi0;
      srcOut[node * 4 + 1] = g * NN + bi1;
      srcOut[node * 4 + 2] = g * NN + bi2;
      srcOut[node * 4 + 3] = g * NN + bi3;
    }
  }
}

// ---------------------------------------------------------------------------
// EdgeConv: block = 16 dst nodes = 64 edges (edges sorted by dst, 4/node).
// Weights are staged transposed into LDS so B-frags are single ds_load_b64s.
// A wave's two 16x16 tiles share row0: one A-frag array feeds two interleaved
// independent WMMA chains. K=128 is chunked at 64 to bound VGPR pressure.
// ---------------------------------------------------------------------------
struct ConvParams {
  const float *w1, *b1, *g1, *be1, *m1, *v1;
  const float *w2, *b2, *g2, *be2, *m2, *v2;
  const float *w3, *b3, *g3, *be3, *m3, *v3;
  const float *ws, *bs, *gs, *bes, *ms, *vs;
};

template <int KD>
__device__ __forceinline__ void stage_wt(const float* __restrict__ W,
                                         float* Wt, int ldw) {
  // Wt[n][k] = W[k][n]; coalesced global reads, 2-way LDS write conflicts max
  for (int idx = threadIdx.x; idx < KD * HID; idx += 256) {
    const int ch = idx & 63;
    const int k  = idx >> 6;
    Wt[ch * ldw + k] = W[k * HID + ch];
  }
}

template <int KD>
__device__ __forceinline__ void gemm_layer(const float* Alds, int lda,
                                           const float* Wt, int ldw,
                                           const float* __restrict__ bias,
                                           const float* __restrict__ bg,
                                           const float* __restrict__ bb,
                                           const float* __restrict__ bm,
                                           const float* __restrict__ bv,
                                           float* Olds, int ldo) {
  constexpr int CH = (KD < 64) ? KD : 64;
  constexpr int NK = CH / 4;
  const int lane = threadIdx.x & 31;
  const int wv   = threadIdx.x >> 5;
  const int lh   = lane >> 4;
  const int lid  = lane & 15;
  const int t0   = wv * 2;                // wave's two tiles share row0
  const int row0 = (t0 >> 2) * 16;
  const int c0   = (t0 & 3) * 16;
  const int c1   = c0 + 16;
  v8f acc0 = {}, acc1 = {};
#pragma unroll
  for (int kc = 0; kc < KD; kc += CH) {
    v2f afr[NK], b0f[NK], b1f[NK];
#pragma unroll
    for (int k4 = 0; k4 < NK; ++k4) {
      const int kb = kc + 4 * k4 + 2 * lh;
      afr[k4] = *(const v2f*)(Alds + (row0 + lid) * lda + kb);
      b0f[k4] = *(const v2f*)(Wt + (c0 + lid) * ldw + kb);
      b1f[k4] = *(const v2f*)(Wt + (c1 + lid) * ldw + kb);
    }
#pragma unroll
    for (int k4 = 0; k4 < NK; ++k4) {     // two independent chains interleave
      acc0 = wmma4(afr[k4], b0f[k4], acc0);
      acc1 = wmma4(afr[k4], b1f[k4], acc1);
    }
  }
  {
    const int ch = c0 + lid;
    const float sc = bg[ch] * rsqrtf(bv[ch] + EPS);
    const float sh = bb[ch] - bm[ch] * sc;
    const float bi = bias[ch];
#pragma unroll
    for (int v = 0; v < 8; ++v) {
      float z = acc0[v] + bi;
      z = (z >= 0.f) ? z : 0.01f * z;
      Olds[(row0 + v + 8 * lh) * ldo + ch] = z * sc + sh;
    }
  }
  {
    const int ch = c1 + lid;
    const float sc = bg[ch] * rsqrtf(bv[ch] + EPS);
    const float sh = bb[ch] - bm[ch] * sc;
    const float bi = bias[ch];
#pragma unroll
    for (int v = 0; v < 8; ++v) {
      float z = acc1[v] + bi;
      z = (z >= 0.f) ? z : 0.01f * z;
      Olds[(row0 + v + 8 * lh) * ldo + ch] = z * sc + sh;
    }
  }
}

template <int CIN>
__global__ __launch_bounds__(256)
void edgeconv_k(const float* __restrict__ Xin, const int* __restrict__ src,
                ConvParams p, float* __restrict__ Cout) {
  constexpr int LDE  = 2 * CIN + 4;       // even + 4-bank row skew
  constexpr int LDW1 = 2 * CIN + 2;       // transposed layer-1 weights
  constexpr int LDWM = (LDW1 > 66) ? LDW1 : 66;
  extern __shared__ float smem[];
  float* E0   = smem;                     // [64][LDE] edge features
  float* act1 = E0 + 64 * LDE;            // [64][68]
  float* act2 = act1 + 64 * 68;           // [64][68]
  float* Wt   = act2 + 64 * 68;           // [64][LDWM] staged transposed W
  int*   slds = (int*)(Wt + 64 * LDWM);   // [64] src node ids
  const int nodeBase = blockIdx.x * 16;
  const int eBase    = nodeBase * 4;

  if (threadIdx.x < 64) slds[threadIdx.x] = src[eBase + threadIdx.x];
  stage_wt<2 * CIN>(p.w1, Wt, LDW1);
  __syncthreads();

  // Build [xi, xj - xi] tile
  for (int idx = threadIdx.x; idx < 64 * 2 * CIN; idx += 256) {
    const int e = idx / (2 * CIN);
    const int j = idx - e * (2 * CIN);
    const int dn = nodeBase + (e >> 2);
    float vv;
    if (j < CIN) {
      vv = Xin[(size_t)dn * CIN + j];
    } else {
      const int jj = j - CIN;
      vv = Xin[(size_t)slds[e] * CIN + jj] - Xin[(size_t)dn * CIN + jj];
    }
    E0[e * LDE + j] = vv;
  }
  __syncthreads();
  gemm_layer<2 * CIN>(E0, LDE, Wt, LDW1, p.b1, p.g1, p.be1, p.m1, p.v1, act1, 68);
  __syncthreads();
  stage_wt<64>(p.w2, Wt, 66);
  __syncthreads();
  gemm_layer<64>(act1, 68, Wt, 66, p.b2, p.g2, p.be2, p.m2, p.v2, act2, 68);
  __syncthreads();
  stage_wt<64>(p.w3, Wt, 66);
  __syncthreads();
  gemm_layer<64>(act2, 68, Wt, 66, p.b3, p.g3, p.be3, p.m3, p.v3, act1, 68);
  __syncthreads();

  // mean over the 4 consecutive edges of each dst + BN(Linear) shortcut
  for (int idx = threadIdx.x; idx < 16 * HID; idx += 256) {
    const int node = idx >> 6;
    const int ch   = idx & 63;
    const float* mr = act1 + (node * 4) * 68 + ch;
    const float agg = 0.25f * (mr[0] + mr[68] + mr[136] + mr[204]);
    const int gn = nodeBase + node;
    float s = p.bs[ch];
    for (int k2 = 0; k2 < CIN; ++k2)
      s += Xin[(size_t)gn * CIN + k2] * p.ws[k2 * HID + ch];
    const float sc = p.gs[ch] * rsqrtf(p.vs[ch] + EPS);
    s = s * sc + (p.bes[ch] - p.ms[ch] * sc);
    Cout[(size_t)gn * HID + ch] = agg + s;
  }
}

// ---------------------------------------------------------------------------
// Global mean pool of one conv output into pooled[:, slot*64 : slot*64+64]
// ---------------------------------------------------------------------------
__global__ void pool_k(const float* __restrict__ C, float* __restrict__ pooled,
                       int slot) {
  const int g = blockIdx.x, f = threadIdx.x;  // 64 threads
  const float* cg = C + (size_t)g * NN * HID;
  float s = 0.f;
  for (int n = 0; n < NN; ++n) s += cg[n * HID + f];
  pooled[g * 192 + slot * 64 + f] = s * (1.f / NN);
}

// ---------------------------------------------------------------------------
// Dense head: bn0 -> d1+leaky+bn1 -> d2+leaky+bn2 -> out -> softmax
// ---------------------------------------------------------------------------
struct HeadParams {
  const float *bn0g, *bn0b, *bn0m, *bn0v;
  const float *w1, *b1;
  const float *bn1g, *bn1b, *bn1m, *bn1v;
  const float *w2, *b2;
  const float *bn2g, *bn2b, *bn2m, *bn2v;
  const float *wo, *bo;
};

__global__ void head_k(const float* __restrict__ pooled,
                       const float* __restrict__ gin, HeadParams hp,
                       float* __restrict__ out) {
  __shared__ float h[196], h2[64], h3[64], lg[2];
  const int g = blockIdx.x, t = threadIdx.x;  // 64 threads
  for (int f = t; f < 192 + FG; f += 64) {
    const float v = (f < 192) ? pooled[g * 192 + f] : gin[g * FG + (f - 192)];
    const float sc = hp.bn0g[f] * rsqrtf(hp.bn0v[f] + EPS);
    h[f] = v * sc + (hp.bn0b[f] - hp.bn0m[f] * sc);
  }
  __syncthreads();
  {
    float a = hp.b1[t];
    for (int k2 = 0; k2 < 196; ++k2) a += h[k2] * hp.w1[k2 * 64 + t];
    a = (a >= 0.f) ? a : 0.01f * a;
    const float sc = hp.bn1g[t] * rsqrtf(hp.bn1v[t] + EPS);
    h2[t] = a * sc + (hp.bn1b[t] - hp.bn1m[t] * sc);
  }
  __syncthreads();
  {
    float a = hp.b2[t];
    for (int k2 = 0; k2 < 64; ++k2) a += h2[k2] * hp.w2[k2 * 64 + t];
    a = (a >= 0.f) ? a : 0.01f * a;
    const float sc = hp.bn2g[t] * rsqrtf(hp.bn2v[t] + EPS);
    h3[t] = a * sc + (hp.bn2b[t] - hp.bn2m[t] * sc);
  }
  __syncthreads();
  if (t < 2) {
    float a = hp.bo[t];
    for (int k2 = 0; k2 < 64; ++k2) a += h3[k2] * hp.wo[k2 * 2 + t];
    lg[t] = a;
  }
  __syncthreads();
  if (t == 0) {
    const float mx = fmaxf(lg[0], lg[1]);
    const float e0 = __expf(lg[0] - mx), e1 = __expf(lg[1] - mx);
    const float inv = 1.f / (e0 + e1);
    out[g * 2 + 0] = e0 * inv;
    out[g * 2 + 1] = e1 * inv;
  }
}

// ---------------------------------------------------------------------------
// Host side. d_in is the setup_inputs() dict flattened depth-first:
//  0:x 1:edge_index 2:graph_input 3..5:gn0(g,b,alpha)
//  6..29:conv1  30..53:conv2  54..77:conv3   (per conv: 3 mlp blocks of
//     [w,b,bn.g,bn.b,bn.m,bn.v] then short [w,b,bn.g,bn.b,bn.m,bn.v])
//  78..81:bn0  82,83:dense1  84..87:bn1  88,89:dense2  90..93:bn2  94,95:out
// ---------------------------------------------------------------------------
extern "C" void kernel_launch(void* const* d_in, const int* in_sizes, int n_in,
                              void* d_out, int out_size, void* d_ws,
                              size_t ws_size, hipStream_t stream) {
  (void)in_sizes; (void)n_in; (void)out_size; (void)ws_size;
  const float* x    = (const float*)d_in[0];
  const int*   edge = (const int*)d_in[1];    // [2,E]; first E = src
  const float* gin  = (const float*)d_in[2];
  const float* gn0g = (const float*)d_in[3];
  const float* gn0b = (const float*)d_in[4];
  const float* gn0a = (const float*)d_in[5];

  auto conv = [&](int base) {
    ConvParams p;
    const float** f = (const float**)&p;
    for (int i = 0; i < 24; ++i) f[i] = (const float*)d_in[base + i];
    return p;
  };
  ConvParams c1p = conv(6), c2p = conv(30), c3p = conv(54);
  HeadParams hp;
  {
    const float** f = (const float**)&hp;
    for (int i = 0; i < 18; ++i) f[i] = (const float*)d_in[78 + i];
  }

  // workspace layout (bytes)
  char* ws = (char*)d_ws;
  float* xn     = (float*)ws;                                  //  8 MB
  float* cA     = (float*)(ws + 8388608);                      // 32 MB
  float* cB     = (float*)(ws + 8388608 + 33554432);           // 32 MB
  int*   srcW   = (int*)  (ws + 8388608 + 67108864);           //  2 MB
  float* pooled = (float*)(ws + 8388608 + 67108864 + 2097152); // 384 KB

  const size_t smConv1 = (size_t)(64 * 36 + 2 * 64 * 68 + 64 * 66) * 4 + 256;
  const size_t smConv2 = (size_t)(64 * 132 + 2 * 64 * 68 + 64 * 130) * 4 + 256;
  const size_t smKnn   = (size_t)(NN * 68 + 4 * 16 * 260 + NN) * 4;

  graphnorm_k<<<NB, 32, 0, stream>>>(x, gn0g, gn0b, gn0a, xn);
  edgeconv_k<16><<<NNODES / 16, 256, smConv1, stream>>>(xn, edge, c1p, cA);
  pool_k<<<NB, 64, 0, stream>>>(cA, pooled, 0);
  knn_k<<<NB * 2, 128, smKnn, stream>>>(cA, srcW);
  edgeconv_k<64><<<NNODES / 16, 256, smConv2, stream>>>(cA, srcW, c2p, cB);
  pool_k<<<NB, 64, 0, stream>>>(cB, pooled, 1);
  knn_k<<<NB * 2, 128, smKnn, stream>>>(cB, srcW);
  edgeconv_k<64><<<NNODES / 16, 256, smConv2, stream>>>(cB, srcW, c3p, cA);
  pool_k<<<NB, 64, 0, stream>>>(cA, pooled, 2);
  head_k<<<NB, 64, 0, stream>>>(pooled, gin, hp, (float*)d_out);
}